// DualFocalLossAblation1_22574348108424
// MI455X (gfx1250) — compile-verified
//
#include <hip/hip_runtime.h>

#define N_COLS 1000
#define WPB 8                        /* waves per block (256 threads) */
#define BLOCK_THREADS (WPB * 32)
#define NUM_BLOCKS 1024
#define N_ROWS 65536
#define ROWS_PER_WAVE (N_ROWS / (NUM_BLOCKS * WPB))   /* = 8 */
#define BUF_FLOATS 1024              /* 4096 B per LDS row buffer */

#define NEG_INF (-__builtin_inff())

// ---------------------------------------------------------------------------
// CDNA5 async global->LDS staging (gfx1250).  offset: immediate applies to
// both the LDS and the global address, so one VGPR pair + one LDS-offset VGPR
// covers a whole 4000-B row in 8 b128 transfers (7 full + 1 for 26 lanes).
// ---------------------------------------------------------------------------
__device__ __forceinline__ void async_load_row(const float* base,
                                               unsigned lds_buf_off, int lane) {
  unsigned long long ga = (unsigned long long)base + (unsigned)(lane * 16);
  unsigned lo = lds_buf_off + (unsigned)(lane * 16);
  asm volatile("global_load_async_to_lds_b128 %0, %1, off"             :: "v"(lo), "v"(ga) : "memory");
  asm volatile("global_load_async_to_lds_b128 %0, %1, off offset:512"  :: "v"(lo), "v"(ga) : "memory");
  asm volatile("global_load_async_to_lds_b128 %0, %1, off offset:1024" :: "v"(lo), "v"(ga) : "memory");
  asm volatile("global_load_async_to_lds_b128 %0, %1, off offset:1536" :: "v"(lo), "v"(ga) : "memory");
  asm volatile("global_load_async_to_lds_b128 %0, %1, off offset:2048" :: "v"(lo), "v"(ga) : "memory");
  asm volatile("global_load_async_to_lds_b128 %0, %1, off offset:2560" :: "v"(lo), "v"(ga) : "memory");
  asm volatile("global_load_async_to_lds_b128 %0, %1, off offset:3072" :: "v"(lo), "v"(ga) : "memory");
  if (lane < 26)  /* vec4 indices 224..249 -> last 416 B */
    asm volatile("global_load_async_to_lds_b128 %0, %1, off offset:3584" :: "v"(lo), "v"(ga) : "memory");
}

#define WAIT_ASYNC_LE8() asm volatile("s_wait_asynccnt 8" ::: "memory")
#define WAIT_ASYNC_LE0() asm volatile("s_wait_asynccnt 0" ::: "memory")
#define WAIT_DS_0()      asm volatile("s_wait_dscnt 0"    ::: "memory")

// ---------------------------------------------------------------------------
// Per-row focal-loss term; all lanes return the same value (butterflies).
// Pads (floats 1000..1023 of each buffer) are pre-set to -inf, which is
// neutral for max, contributes exp(-inf)=0 to the sum, and yields p=0 when
// fewer than 2 masked candidates exist (matching top_k zero-fill).
// ---------------------------------------------------------------------------
__device__ __forceinline__ float compute_row(const float4* buf, float xk) {
  // pass 1: row max
  float m = NEG_INF;
#pragma unroll
  for (int j = 0; j < 8; ++j) {
    float a = fmaxf(fmaxf(buf[j].x, buf[j].y), fmaxf(buf[j].z, buf[j].w));
    m = fmaxf(m, a);
  }
#pragma unroll
  for (int o = 16; o > 0; o >>= 1)
    m = fmaxf(m, __shfl_xor(m, o, 32));

  // pass 2: exp-sum + masked top-2 (4 independent chains)
  const float L2E = 1.4426950408889634f;   // log2(e)
  const float nm  = -m * L2E;
  float s0 = 0.f, s1 = 0.f, s2 = 0.f, s3 = 0.f;
  float t1a = NEG_INF, t2a = NEG_INF, t1b = NEG_INF, t2b = NEG_INF;
  float t1c = NEG_INF, t2c = NEG_INF, t1d = NEG_INF, t2d = NEG_INF;
#pragma unroll
  for (int j = 0; j < 8; ++j) {
    float a = buf[j].x, b = buf[j].y, c = buf[j].z, d = buf[j].w;
    s0 += exp2f(__fmaf_rn(a, L2E, nm));
    s1 += exp2f(__fmaf_rn(b, L2E, nm));
    s2 += exp2f(__fmaf_rn(c, L2E, nm));
    s3 += exp2f(__fmaf_rn(d, L2E, nm));
    float am = (a < xk) ? a : NEG_INF;
    float bm = (b < xk) ? b : NEG_INF;
    float cm = (c < xk) ? c : NEG_INF;
    float dm = (d < xk) ? d : NEG_INF;
    t2a = fmaxf(t2a, fminf(t1a, am)); t1a = fmaxf(t1a, am);
    t2b = fmaxf(t2b, fminf(t1b, bm)); t1b = fmaxf(t1b, bm);
    t2c = fmaxf(t2c, fminf(t1c, cm)); t1c = fmaxf(t1c, cm);
    t2d = fmaxf(t2d, fminf(t1d, dm)); t1d = fmaxf(t1d, dm);
  }
  float s = (s0 + s1) + (s2 + s3);

  float t1 = fmaxf(t1a, t1b);
  float t2 = fmaxf(fminf(t1a, t1b), fmaxf(t2a, t2b));
  float u1 = fmaxf(t1c, t1d);
  float u2 = fmaxf(fminf(t1c, t1d), fmaxf(t2c, t2d));
  t2 = fmaxf(fminf(t1, u1), fmaxf(t2, u2));
  t1 = fmaxf(t1, u1);

#pragma unroll
  for (int o = 16; o > 0; o >>= 1) {
    s += __shfl_xor(s, o, 32);
    float o1 = __shfl_xor(t1, o, 32);
    float o2 = __shfl_xor(t2, o, 32);
    t2 = fmaxf(fminf(t1, o1), fmaxf(t2, o2));
    t1 = fmaxf(t1, o1);
  }

  float log2S = __log2f(s);
  float zk    = __fmaf_rn(xk, L2E, nm);
  float logpk = (zk - log2S) * 0.6931471805599453f;  // ln(p_k)
  float pk = exp2f(zk - log2S);
  float pj = exp2f(__fmaf_rn(t1, L2E, nm) - log2S);  // 0 if t1 == -inf
  float p3 = exp2f(__fmaf_rn(t2, L2E, nm) - log2S);  // 0 if t2 == -inf
  float u = 1.0f - pk + pj + p3;
  return -(u * u) * logpk;
}

// ---------------------------------------------------------------------------
// One wave per row, 8 rows per wave, async double buffer in LDS.
// ---------------------------------------------------------------------------
__global__ __launch_bounds__(BLOCK_THREADS)
void dual_focal_loss_kernel(const float* __restrict__ x,
                            const int* __restrict__ target,
                            float* __restrict__ out) {
  __shared__ __align__(16) float tile[WPB * 2 * BUF_FLOATS];   // 64 KB
  __shared__ float part[WPB];

  const int lane = threadIdx.x & 31;
  const int wv   = threadIdx.x >> 5;
  const int gw   = blockIdx.x * WPB + wv;
  const int row0 = gw * ROWS_PER_WAVE;

  // LDS byte offset of this wave's two buffers (for the raw async asm)
  typedef __attribute__((address_space(3))) float lds_f;
  unsigned tile_off = (unsigned)(unsigned long long)(lds_f*)tile;
  unsigned mybuf0   = tile_off + (unsigned)(wv * 2 * BUF_FLOATS * 4);

  // fill the 96-B pad of both buffers with -inf once (async never writes it)
  if (lane < 24) {
    tile[(wv * 2 + 0) * BUF_FLOATS + 1000 + lane] = NEG_INF;
    tile[(wv * 2 + 1) * BUF_FLOATS + 1000 + lane] = NEG_INF;
  }

  // wave-uniform target indices -> scalar loads
  int tg[ROWS_PER_WAVE];
#pragma unroll
  for (int i = 0; i < ROWS_PER_WAVE; ++i) tg[i] = target[row0 + i];

  // prime the pipeline: row 0 -> buffer 0
  async_load_row(x + (size_t)row0 * N_COLS, mybuf0, lane);

  float acc = 0.f;
#pragma unroll
  for (int i = 0; i < ROWS_PER_WAVE; ++i) {
    const float* base = x + (size_t)(row0 + i) * N_COLS;
    if (i + 1 < ROWS_PER_WAVE) {
      WAIT_DS_0();   // prior reads of the destination buffer have retired
      async_load_row(x + (size_t)(row0 + i + 1) * N_COLS,
                     mybuf0 + (unsigned)(((i + 1) & 1) * (BUF_FLOATS * 4)), lane);
    }
    if (i + 2 < ROWS_PER_WAVE)
      __builtin_prefetch(x + (size_t)(row0 + i + 2) * N_COLS + lane * 32, 0, 3);

    if (i + 1 < ROWS_PER_WAVE) WAIT_ASYNC_LE8(); else WAIT_ASYNC_LE0();

    // LDS -> registers (8 unconditional ds_load_b128; pad is -inf)
    const float4* lp =
        (const float4*)&tile[(wv * 2 + (i & 1)) * BUF_FLOATS];
    float4 buf[8];
#pragma unroll
    for (int j = 0; j < 8; ++j) buf[j] = lp[lane + 32 * j];

    float xk = base[tg[i]];                  // wave-uniform scalar load
    acc += compute_row(buf, xk);
  }

  // block reduction: 8 wave partials -> 1 atomic per block
  if (lane == 0) part[wv] = acc;
  __syncthreads();
  if (threadIdx.x == 0) {
    float s = 0.f;
#pragma unroll
    for (int j = 0; j < WPB; ++j) s += part[j];
    atomicAdd(out, s);
  }
}

__global__ void zero_out_kernel(float* out) { out[0] = 0.0f; }

// ---------------------------------------------------------------------------
extern "C" void kernel_launch(void* const* d_in, const int* in_sizes, int n_in,
                              void* d_out, int out_size, void* d_ws, size_t ws_size,
                              hipStream_t stream) {
  (void)in_sizes; (void)n_in; (void)out_size; (void)d_ws; (void)ws_size;
  const float* x   = (const float*)d_in[0];
  const int*   tgt = (const int*)d_in[1];
  float*       out = (float*)d_out;

  zero_out_kernel<<<1, 1, 0, stream>>>(out);
  dual_focal_loss_kernel<<<NUM_BLOCKS, BLOCK_THREADS, 0, stream>>>(x, tgt, out);
}